// YOLOv2Loss_17282948399386
// MI455X (gfx1250) — compile-verified
//
#include <hip/hip_runtime.h>
#include <hip/hip_bf16.h>
#include <stdint.h>

// ---- problem constants (match reference) ----
#define AN     5
#define NCLS   20
#define NCH    25          // 5 + NC
#define BN     32
#define FN     38
#define SN     (FN * FN)   // 1444
#define ON     50
#define SAN    (SN * AN)   // 7220
#define THRESH 0.6f
#define OBJ_SCALE 5.0f

typedef __attribute__((ext_vector_type(2))) float v2f;
typedef __attribute__((ext_vector_type(8))) float v8f;

// CDNA5 async global->LDS path (probe-verified live on this toolchain).
#if defined(__has_builtin)
#  if __has_builtin(__builtin_amdgcn_global_load_async_to_lds_b32) && \
      __has_builtin(__builtin_amdgcn_s_wait_asynccnt)
#    define USE_ASYNC_LDS 1
#  endif
#endif
#ifndef USE_ASYNC_LDS
#  define USE_ASYNC_LDS 0
#endif

#if USE_ASYNC_LDS
typedef __attribute__((address_space(1))) int g_i32;   // "__device__ int *"
typedef __attribute__((address_space(3))) int l_i32;   // "__shared__ int *"
#endif

struct Rec {
  int valid, cell, anc, cls;
  float d0, d1, d2, d3, mi;
  int pad;
};

__device__ __forceinline__ float fast_rcp(float x) {
  return __builtin_amdgcn_rcpf(x);     // v_rcp_f32, ~1 ulp: fine for loss math
}
__device__ __forceinline__ float sigmoidf(float x) {
  return fast_rcp(1.0f + __expf(-x));
}

__device__ __forceinline__ float iou_xywh(float px, float py, float pw, float ph,
                                          float gx, float gy, float gw, float gh) {
  float tlx = fmaxf(px - pw * 0.5f, gx - gw * 0.5f);
  float tly = fmaxf(py - ph * 0.5f, gy - gh * 0.5f);
  float brx = fminf(px + pw * 0.5f, gx + gw * 0.5f);
  float bry = fminf(py + ph * 0.5f, gy + gh * 0.5f);
  float iw = fmaxf(brx - tlx, 0.0f);
  float ih = fmaxf(bry - tly, 0.0f);
  float inter = iw * ih;
  return inter * fast_rcp(pw * ph + gw * gh - inter + 1e-12f);
}

// Block-wide (256 threads) sum -> atomicAdd; 64->1 stage on the matrix pipe:
// A(16x4) = partials, B(4x16) = ones  =>  D rows are row-sums.
__device__ __forceinline__ void block_sum_atomic(float v, float* gout, float* sred) {
  sred[threadIdx.x] = v;
  __syncthreads();
  if (threadIdx.x < 32) {                 // wave 0, EXEC all-ones (WMMA req.)
    int lane = threadIdx.x;
    int base = lane * 8;
    v2f a;
    a.x = sred[base + 0] + sred[base + 1] + sred[base + 2] + sred[base + 3];
    a.y = sred[base + 4] + sred[base + 5] + sred[base + 6] + sred[base + 7];
    v2f b; b.x = 1.0f; b.y = 1.0f;
    v8f c = {};
    c = __builtin_amdgcn_wmma_f32_16x16x4_f32(false, a, false, b,
                                              (short)0, c, false, false);
    float s = c[0] + c[1] + c[2] + c[3] + c[4] + c[5] + c[6] + c[7];
    // lanes 0..15 hold sum of rows 0..7; lanes 16..31 hold sum of rows 8..15
    float tot = __shfl(s, 0, 32) + __shfl(s, 16, 32);
    if (lane == 0) atomicAdd(gout, tot);
  }
  __syncthreads();
}

__global__ void k_init(float* out, int* npos) {
  if (blockIdx.x == 0) {
    if (threadIdx.x == 0) out[0] = 0.0f;
    if (threadIdx.x < BN) npos[threadIdx.x] = 0;
  }
}

// grid = (ceil(SAN/256), BN), block = 256
__global__ void k_maxiou(const float* __restrict__ outputs,
                         const float* __restrict__ targets,
                         const float* __restrict__ anchors,
                         float* __restrict__ maxiou,
                         int* __restrict__ npos) {
  __shared__ float s_t[ON * 5];        // raw targets (async-staged)
  __shared__ float s_box[ON * 5];      // corner form: x1,y1,x2,y2,area
  __shared__ float s_anc[2 * AN];
  int b = blockIdx.y;
  const float* tsrc = targets + (size_t)b * ON * 5;

#if USE_ASYNC_LDS
  if (threadIdx.x < ON * 5) {
    __builtin_amdgcn_global_load_async_to_lds_b32(
        (g_i32*)(const_cast<float*>(tsrc) + threadIdx.x),
        (l_i32*)&s_t[threadIdx.x], 0, 0);
  }
  if (threadIdx.x < 2 * AN) s_anc[threadIdx.x] = anchors[threadIdx.x];
  __builtin_amdgcn_s_wait_asynccnt(0);
  __syncthreads();
#else
  if (threadIdx.x < ON * 5) s_t[threadIdx.x] = tsrc[threadIdx.x];
  if (threadIdx.x < 2 * AN) s_anc[threadIdx.x] = anchors[threadIdx.x];
  __syncthreads();
#endif

  // Once per block: gt boxes -> corner form, invalid -> degenerate box
  // (x1,y1=+BIG, x2,y2=-BIG, area=0) so the IoU is exactly 0 branchlessly.
  if (threadIdx.x < ON) {
    int o = threadIdx.x;
    float tx = s_t[o * 5 + 0], ty = s_t[o * 5 + 1];
    float tw = s_t[o * 5 + 2], th = s_t[o * 5 + 3], tc = s_t[o * 5 + 4];
    bool valid = (tx + ty + tw + th + tc) > 0.0f;
    float gx = tx * FN, gy = ty * FN, gw = tw * FN, gh = th * FN;
    const float BIG = 1e30f;
    s_box[o * 5 + 0] = valid ? gx - gw * 0.5f : BIG;
    s_box[o * 5 + 1] = valid ? gy - gh * 0.5f : BIG;
    s_box[o * 5 + 2] = valid ? gx + gw * 0.5f : -BIG;
    s_box[o * 5 + 3] = valid ? gy + gh * 0.5f : -BIG;
    s_box[o * 5 + 4] = valid ? gw * gh : 0.0f;
  }
  __syncthreads();

  int r = blockIdx.x * blockDim.x + threadIdx.x;
  float m = 0.0f;
  bool inb = (r < SAN);
  if (inb) {
    int cell = r / AN, a = r % AN;
    int i = cell / FN, j = cell % FN;
    size_t obase = ((size_t)b * AN + a) * NCH * (size_t)SN + (size_t)cell;
    float t0 = outputs[obase + 0 * SN];
    float t1 = outputs[obase + 1 * SN];
    float t2 = outputs[obase + 2 * SN];
    float t3 = outputs[obase + 3 * SN];
    float px = sigmoidf(t0) + (float)j;
    float py = sigmoidf(t1) + (float)i;
    float pw = __expf(t2) * s_anc[2 * a + 0] * (float)FN;
    float ph = __expf(t3) * s_anc[2 * a + 1] * (float)FN;
    float px1 = px - pw * 0.5f, py1 = py - ph * 0.5f;
    float px2 = px + pw * 0.5f, py2 = py + ph * 0.5f;
    float parea = pw * ph;
#pragma unroll 5
    for (int o = 0; o < ON; ++o) {
      float gx1 = s_box[o * 5 + 0], gy1 = s_box[o * 5 + 1];
      float gx2 = s_box[o * 5 + 2], gy2 = s_box[o * 5 + 3];
      float ga  = s_box[o * 5 + 4];
      float iw = fmaxf(fminf(px2, gx2) - fmaxf(px1, gx1), 0.0f);
      float ih = fmaxf(fminf(py2, gy2) - fmaxf(py1, gy1), 0.0f);
      float inter = iw * ih;
      float iou = inter * fast_rcp(parea + ga - inter + 1e-12f);
      m = fmaxf(m, iou);
    }
    maxiou[(size_t)b * SAN + r] = m;
  }
  unsigned long long bal = __ballot(inb && (m > THRESH));
  if (bal != 0ull && (threadIdx.x & 31u) == 0u) atomicOr(&npos[b], 1);
}

// grid = BN blocks, block = 64
__global__ void k_assign(const float* __restrict__ targets,
                         const float* __restrict__ anchors,
                         const float* __restrict__ maxiou,
                         Rec* __restrict__ recs) {
  __shared__ int s_key[ON];
  int b = blockIdx.x;
  int o = threadIdx.x;
  int valid = 0, cell = 0, anc = 0, cls = 0;
  float d0 = 0, d1 = 0, d2 = 0, d3 = 0, mi = 0;
  if (o < ON) {
    const float* t = targets + ((size_t)b * ON + o) * 5;
    float tx = t[0], ty = t[1], tw = t[2], th = t[3], tc = t[4];
    valid = (tx + ty + tw + th + tc > 0.0f) ? 1 : 0;
    float gx = tx * FN, gy = ty * FN, gw = tw * FN, gh = th * FN;
    int cx = (int)floorf(gx - gw * 0.5f);
    int cy = (int)floorf(gy - gh * 0.5f);
    int c = cy * FN + cx;
    c = min(max(c, 0), SN - 1);
    cell = c;
    float ax = (float)(c % FN) + 0.5f;
    float ay = (float)(c / FN) + 0.5f;
    float best = -1.0f;
#pragma unroll
    for (int a = 0; a < AN; ++a) {
      float aw = anchors[2 * a] * FN, ah = anchors[2 * a + 1] * FN;
      float ov = iou_xywh(ax, ay, aw, ah, gx, gy, gw, gh);
      if (ov > best) { best = ov; anc = a; }   // first-max wins (argmax)
    }
    float aw = anchors[2 * anc] * FN, ah = anchors[2 * anc + 1] * FN;
    d0 = gx - ax; d1 = gy - ay; d2 = gw / aw; d3 = gh / ah;
    cls = (int)tc;
    cls = min(max(cls, 0), NCLS - 1);
    mi = maxiou[(size_t)b * SAN + (size_t)cell * AN + anc];
    s_key[o] = valid ? (cell * AN + anc) : (-1 - o);
  }
  __syncthreads();
  if (o < ON && valid) {
    int key = s_key[o];
    for (int o2 = o + 1; o2 < ON; ++o2)
      if (s_key[o2] == key) { valid = 0; break; }   // last writer wins
  }
  if (o < ON) {
    Rec r;
    r.valid = valid; r.cell = cell; r.anc = anc; r.cls = cls;
    r.d0 = d0; r.d1 = d1; r.d2 = d2; r.d3 = d3; r.mi = mi; r.pad = 0;
    recs[(size_t)b * ON + o] = r;
  }
}

// grid = (ceil(SAN/256), BN), block = 256
__global__ void k_noobj(const float* __restrict__ outputs,
                        const float* __restrict__ maxiou,
                        const int* __restrict__ npos,
                        float* __restrict__ out) {
  __shared__ float sred[256];
  int b = blockIdx.y;
  int r = blockIdx.x * blockDim.x + threadIdx.x;
  float contrib = 0.0f;
  if (r < SAN) {
    int cell = r / AN, a = r % AN;
    float logit = outputs[(((size_t)b * AN + a) * NCH + 4) * (size_t)SN + cell];
    float conf = sigmoidf(logit);
    float m = maxiou[(size_t)b * SAN + r];
    float im = (npos[b] && (m >= THRESH)) ? 0.0f : 1.0f;  // NOOBJ_SCALE = 1
    float t = conf * im;
    contrib = t * t;
  }
  block_sum_atomic(contrib, out, sred);
}

// grid = ceil(BN*ON/256), block = 256
__global__ void k_resp(const float* __restrict__ outputs,
                       const float* __restrict__ maxiou,
                       const int* __restrict__ npos,
                       const Rec* __restrict__ recs,
                       float* __restrict__ out) {
  __shared__ float sred[256];
  int idx = blockIdx.x * blockDim.x + threadIdx.x;
  float contrib = 0.0f;
  if (idx < BN * ON) {
    Rec rec = recs[idx];
    if (rec.valid) {
      int b = idx / ON;
      size_t obase = ((size_t)b * AN + rec.anc) * NCH * (size_t)SN + (size_t)rec.cell;
      float t0 = outputs[obase + 0 * SN];
      float t1 = outputs[obase + 1 * SN];
      float t2 = outputs[obase + 2 * SN];
      float t3 = outputs[obase + 3 * SN];
      float t4 = outputs[obase + 4 * SN];
      // box loss (COORD_SCALE = 1)
      float e0 = sigmoidf(t0) - rec.d0;
      float e1 = sigmoidf(t1) - rec.d1;
      float e2 = __expf(t2)  - rec.d2;
      float e3 = __expf(t3)  - rec.d3;
      float box = e0 * e0 + e1 * e1 + e2 * e2 + e3 * e3;
      // obj conf loss: replace the base (noobj) term already accumulated
      float conf = sigmoidf(t4);
      float m = maxiou[(size_t)b * SAN + (size_t)rec.cell * AN + rec.anc];
      float imb = (npos[b] && (m >= THRESH)) ? 0.0f : 1.0f;
      float base = conf * imb; base = base * base;
      float diff = OBJ_SCALE * (conf - rec.mi);
      float il = diff * diff - base;
      // class CE: softmax then log_softmax of probs (matches reference)
      float lg[NCLS];
      float lmax = -1e30f;
#pragma unroll
      for (int k = 0; k < NCLS; ++k) {
        lg[k] = outputs[obase + (size_t)(5 + k) * SN];
        lmax = fmaxf(lmax, lg[k]);
      }
      float Z = 0.0f;
#pragma unroll
      for (int k = 0; k < NCLS; ++k) { lg[k] = __expf(lg[k] - lmax); Z += lg[k]; }
      float rZ = fast_rcp(Z);
      float sumexp = 0.0f, pct = 0.0f;
#pragma unroll
      for (int k = 0; k < NCLS; ++k) {
        float p = lg[k] * rZ;            // probs in (0,1): exp() is safe
        sumexp += __expf(p);
        if (k == rec.cls) pct = p;
      }
      float ce = __logf(sumexp) - pct;
      contrib = box + il + ce;
    }
  }
  block_sum_atomic(contrib, out, sred);
}

__global__ void k_scale(float* out) {
  if (threadIdx.x == 0 && blockIdx.x == 0) out[0] *= (1.0f / (float)BN);
}

extern "C" void kernel_launch(void* const* d_in, const int* in_sizes, int n_in,
                              void* d_out, int out_size, void* d_ws, size_t ws_size,
                              hipStream_t stream) {
  const float* outputs = (const float*)d_in[0];   // (32,125,38,38)
  const float* targets = (const float*)d_in[1];   // (32,50,5)
  const float* anchors = (const float*)d_in[2];   // (5,2)
  float* out = (float*)d_out;

  char* ws = (char*)d_ws;
  size_t off = 0;
  float* maxiou = (float*)(ws + off); off += sizeof(float) * (size_t)BN * SAN;
  int* npos = (int*)(ws + off);       off += 256;   // 32 ints, padded
  Rec* recs = (Rec*)(ws + off);

  k_init<<<1, 64, 0, stream>>>(out, npos);

  dim3 g1((SAN + 255) / 256, BN);
  k_maxiou<<<g1, 256, 0, stream>>>(outputs, targets, anchors, maxiou, npos);
  k_assign<<<BN, 64, 0, stream>>>(targets, anchors, maxiou, recs);
  k_noobj<<<g1, 256, 0, stream>>>(outputs, maxiou, npos, out);
  k_resp<<<(BN * ON + 255) / 256, 256, 0, stream>>>(outputs, maxiou, npos, recs, out);
  k_scale<<<1, 32, 0, stream>>>(out);
}